// GAT_39556648796773
// MI455X (gfx1250) — compile-verified
//
#include <hip/hip_runtime.h>
#include <hip/hip_bf16.h>

// ---------------------------------------------------------------------------
// GAT forward for MI455X (gfx1250, wave32, WMMA).
//   H1 = X @ W1                       (V_WMMA_F32_16X16X4_F32 — exact fp32)
//   hidden = elu(gat_aggregate(H1, a1))
//   H2 = hidden @ W2                  (same WMMA path)
//   out = log_softmax(gat_aggregate(H2, a2))
// Edge phase: ordered-uint atomicMax (segment max over negative-capable
// scores), f32 atomicAdd segment-sum + scatter-add aggregation. H1 (25.6MB)
// and H2 (12.8MB) fit in the 192MB L2, so edge gathers/atomics are
// L2-resident; HBM traffic is ~150MB total (~7us at 23.3TB/s) — the kernel is
// L2-atomic-throughput bound, compute is trivial (3.7 GFLOP).
// ---------------------------------------------------------------------------

#define N_NODES   100000
#define N_EDGES   1600000
#define IN_DIM    256
#define HID_DIM   64
#define NUM_CLS   32
#define NEG_SLOPE 0.01f

typedef __attribute__((ext_vector_type(2))) float v2f;
typedef __attribute__((ext_vector_type(8))) float v8f;

// ---------------- WMMA GEMM: C[nrows,NDIM] = A[nrows,KDIM] @ B[KDIM,NDIM] ---
// One wave32 per 16x16 output tile, full fp32 via V_WMMA_F32_16X16X4_F32.
// Fragment layouts per CDNA5 ISA 7.12.2:
//   A 16x4 f32 : lane(0-15)=row M, VGPR0/1 = K {0,1}; lanes 16-31 = K {2,3}
//   B 4x16 f32 : lane carries column N, VGPRs carry the K rows (symmetric)
//   C/D 16x16  : VGPR i -> row (half*8 + i), lane&15 -> column
template <int KDIM, int NDIM>
__global__ __launch_bounds__(256) void gemm_wmma(const float* __restrict__ A,
                                                 const float* __restrict__ B,
                                                 float* __restrict__ C,
                                                 int nrows) {
  const int lane  = threadIdx.x & 31;
  const int gwave = blockIdx.x * (blockDim.x >> 5) + (threadIdx.x >> 5);
  constexpr int NT = NDIM / 16;
  const int mt = gwave / NT;
  const int nt = gwave % NT;
  if (mt * 16 >= nrows) return;  // wave-uniform: EXEC stays all-ones for WMMA

  const int half = lane >> 4;    // 0 -> K {0,1}; 1 -> K {2,3} within the step
  const int r    = lane & 15;
  const int row  = mt * 16 + r;  // A row carried by this lane
  const int col  = nt * 16 + r;  // B/C column carried by this lane
  const int koff = 2 * half;

  v8f acc = {};
#pragma unroll 4
  for (int k0 = 0; k0 < KDIM; k0 += 4) {
    const float* ap = A + (size_t)row * KDIM + k0 + koff;
    const float* bp = B + (size_t)(k0 + koff) * NDIM + col;
    v2f a, b;
    a[0] = ap[0];
    a[1] = ap[1];
    b[0] = bp[0];
    b[1] = bp[NDIM];
    acc = __builtin_amdgcn_wmma_f32_16x16x4_f32(
        /*neg_a=*/false, a, /*neg_b=*/false, b,
        /*c_mod=*/(short)0, acc, /*reuse_a=*/false, /*reuse_b=*/false);
  }

  const int mbase = mt * 16 + half * 8;
#pragma unroll
  for (int i = 0; i < 8; ++i)
    C[(size_t)(mbase + i) * NDIM + col] = acc[i];
}

// ------- per-node attention scores: s = H@a[:D], d = H@a[D:] ---------------
// One wave per row; lane-strided loads are fully coalesced; shfl reduction.
__global__ __launch_bounds__(256) void node_scores(const float* __restrict__ H,
                                                   const float* __restrict__ a,
                                                   float* __restrict__ s,
                                                   float* __restrict__ d,
                                                   int D, int n) {
  const int lane = threadIdx.x & 31;
  const int row  = blockIdx.x * (blockDim.x >> 5) + (threadIdx.x >> 5);
  if (row >= n) return;  // wave-uniform
  float ss = 0.f, dd = 0.f;
  for (int j = lane; j < D; j += 32) {
    float v = H[(size_t)row * D + j];
    ss = fmaf(v, a[j], ss);
    dd = fmaf(v, a[D + j], dd);
  }
#pragma unroll
  for (int off = 16; off; off >>= 1) {
    ss += __shfl_xor(ss, off, 32);
    dd += __shfl_xor(dd, off, 32);
  }
  if (lane == 0) {
    s[row] = ss;
    d[row] = dd;
  }
}

// ------- order-preserving float<->uint map so atomicMax works on floats -----
__device__ __forceinline__ unsigned ford(float f) {
  unsigned u = __float_as_uint(f);
  return (u & 0x80000000u) ? ~u : (u | 0x80000000u);
}
__device__ __forceinline__ float ford_inv(unsigned v) {
  unsigned u = (v & 0x80000000u) ? (v & 0x7FFFFFFFu) : ~v;
  return __uint_as_float(u);
}

// e = leaky_relu(s[src]+d[dst]); segment max over src (ordered-uint atomicMax)
__global__ __launch_bounds__(256) void edge_score_max(
    const long long* __restrict__ src, const long long* __restrict__ dst,
    const float* __restrict__ s, const float* __restrict__ d,
    float* __restrict__ e, unsigned* __restrict__ mOrd, int nE) {
  int i = blockIdx.x * blockDim.x + threadIdx.x;
  if (i >= nE) return;
  int u = (int)src[i], v = (int)dst[i];
  float x = s[u] + d[v];
  x = (x >= 0.f) ? x : NEG_SLOPE * x;
  e[i] = x;
  atomicMax(&mOrd[u], ford(x));
}

// ex = exp(e - m[src]) (in place); denom[src] += ex
__global__ __launch_bounds__(256) void edge_exp_sum(
    const long long* __restrict__ src, float* __restrict__ e,
    const unsigned* __restrict__ mOrd, float* __restrict__ denom, int nE) {
  int i = blockIdx.x * blockDim.x + threadIdx.x;
  if (i >= nE) return;
  int u = (int)src[i];
  float ex = expf(e[i] - ford_inv(mOrd[u]));
  e[i] = ex;
  atomicAdd(&denom[u], ex);
}

// att = ex / denom[src] (in place) — one divide per edge instead of per
// (edge,feature) in the aggregate kernel.
__global__ __launch_bounds__(256) void edge_att(
    const long long* __restrict__ src, float* __restrict__ e,
    const float* __restrict__ denom, int nE) {
  int i = blockIdx.x * blockDim.x + threadIdx.x;
  if (i >= nE) return;
  e[i] = e[i] / denom[(int)src[i]];
}

// out[src,:] += att * H[dst,:]; one thread per (edge,feature); feature index
// is the fast dimension so H gathers / out atomics are coalesced per edge.
__global__ __launch_bounds__(256) void edge_aggregate(
    const long long* __restrict__ src, const long long* __restrict__ dst,
    const float* __restrict__ att, const float* __restrict__ H,
    float* __restrict__ out, int dshift, long long total) {
  long long t = (long long)blockIdx.x * blockDim.x + threadIdx.x;
  if (t >= total) return;
  int i = (int)(t >> dshift);
  int j = (int)(t & ((1 << dshift) - 1));
  int u = (int)src[i], v = (int)dst[i];
  atomicAdd(&out[((size_t)u << dshift) + j], att[i] * H[((size_t)v << dshift) + j]);
}

__global__ __launch_bounds__(256) void elu_inplace(float* __restrict__ x, long long n) {
  long long i = (long long)blockIdx.x * blockDim.x + threadIdx.x;
  if (i >= n) return;
  float v = x[i];
  x[i] = (v > 0.f) ? v : expm1f(v);
}

// row-wise log_softmax over [n, 32]; one wave per row, lane = class
__global__ __launch_bounds__(256) void log_softmax32(float* __restrict__ x, int n) {
  const int lane = threadIdx.x & 31;
  const int row  = blockIdx.x * (blockDim.x >> 5) + (threadIdx.x >> 5);
  if (row >= n) return;  // wave-uniform
  float v = x[(size_t)row * 32 + lane];
  float m = v;
#pragma unroll
  for (int off = 16; off; off >>= 1) m = fmaxf(m, __shfl_xor(m, off, 32));
  float sum = expf(v - m);
#pragma unroll
  for (int off = 16; off; off >>= 1) sum += __shfl_xor(sum, off, 32);
  x[(size_t)row * 32 + lane] = v - m - logf(sum);
}

// ---------------------------------------------------------------------------
extern "C" void kernel_launch(void* const* d_in, const int* in_sizes, int n_in,
                              void* d_out, int out_size, void* d_ws, size_t ws_size,
                              hipStream_t stream) {
  const float*     X   = (const float*)d_in[0];                 // [N,256]
  const long long* EI  = (const long long*)d_in[1];             // [2,E] int64
  const float*     W1  = (const float*)d_in[2];                 // [256,64]
  const float*     W2  = (const float*)d_in[3];                 // [64,32]
  const float*     A1  = (const float*)d_in[4];                 // [128]
  const float*     A2  = (const float*)d_in[5];                 // [64]
  const long long* SRC = EI;
  const long long* DST = EI + N_EDGES;
  float* OUT = (float*)d_out;                                   // [N,32]

  // ---- carve workspace (256B aligned) ----
  size_t off = 0;
  auto carve = [&](size_t bytes) -> void* {
    void* p = (char*)d_ws + off;
    off += (bytes + 255) & ~(size_t)255;
    return p;
  };
  float*    H1   = (float*)carve((size_t)N_NODES * HID_DIM * 4);
  float*    HID  = (float*)carve((size_t)N_NODES * HID_DIM * 4);  // agg1/hidden
  float*    H2   = (float*)carve((size_t)N_NODES * NUM_CLS * 4);
  float*    s1   = (float*)carve((size_t)N_NODES * 4);
  float*    dd1  = (float*)carve((size_t)N_NODES * 4);
  float*    den1 = (float*)carve((size_t)N_NODES * 4);
  float*    s2   = (float*)carve((size_t)N_NODES * 4);
  float*    dd2  = (float*)carve((size_t)N_NODES * 4);
  float*    den2 = (float*)carve((size_t)N_NODES * 4);
  unsigned* m1   = (unsigned*)carve((size_t)N_NODES * 4);
  unsigned* m2   = (unsigned*)carve((size_t)N_NODES * 4);
  float*    exb  = (float*)carve((size_t)N_EDGES * 4);            // e/ex/att, reused
  (void)ws_size; (void)n_in; (void)in_sizes; (void)out_size;

  const int eBlocks = (N_EDGES + 255) / 256;
  const int wBlocks = (N_NODES + 7) / 8;   // wave-per-row kernels, 8 waves/block

  // ================= layer 1 =================
  // GEMM1: 6250 row-tiles x 4 col-tiles = 25000 waves, 8 waves/block
  gemm_wmma<IN_DIM, HID_DIM><<<25000 / 8, 256, 0, stream>>>(X, W1, H1, N_NODES);
  node_scores<<<wBlocks, 256, 0, stream>>>(H1, A1, s1, dd1, HID_DIM, N_NODES);

  hipMemsetAsync(m1,   0, (size_t)N_NODES * 4, stream);
  hipMemsetAsync(den1, 0, (size_t)N_NODES * 4, stream);
  hipMemsetAsync(HID,  0, (size_t)N_NODES * HID_DIM * 4, stream);

  edge_score_max<<<eBlocks, 256, 0, stream>>>(SRC, DST, s1, dd1, exb, m1, N_EDGES);
  edge_exp_sum<<<eBlocks, 256, 0, stream>>>(SRC, exb, m1, den1, N_EDGES);
  edge_att<<<eBlocks, 256, 0, stream>>>(SRC, exb, den1, N_EDGES);
  {
    long long total = (long long)N_EDGES * HID_DIM;
    edge_aggregate<<<(int)((total + 255) / 256), 256, 0, stream>>>(
        SRC, DST, exb, H1, HID, 6, total);
  }
  {
    long long total = (long long)N_NODES * HID_DIM;
    elu_inplace<<<(int)((total + 255) / 256), 256, 0, stream>>>(HID, total);
  }

  // ================= layer 2 =================
  // GEMM2: 6250 x 2 = 12500 waves -> 1563 blocks (tail waves self-mask uniformly)
  gemm_wmma<HID_DIM, NUM_CLS><<<(12500 + 7) / 8, 256, 0, stream>>>(HID, W2, H2, N_NODES);
  node_scores<<<wBlocks, 256, 0, stream>>>(H2, A2, s2, dd2, NUM_CLS, N_NODES);

  hipMemsetAsync(m2,   0, (size_t)N_NODES * 4, stream);
  hipMemsetAsync(den2, 0, (size_t)N_NODES * 4, stream);
  hipMemsetAsync(OUT,  0, (size_t)N_NODES * NUM_CLS * 4, stream);

  edge_score_max<<<eBlocks, 256, 0, stream>>>(SRC, DST, s2, dd2, exb, m2, N_EDGES);
  edge_exp_sum<<<eBlocks, 256, 0, stream>>>(SRC, exb, m2, den2, N_EDGES);
  edge_att<<<eBlocks, 256, 0, stream>>>(SRC, exb, den2, N_EDGES);
  {
    long long total = (long long)N_EDGES * NUM_CLS;
    edge_aggregate<<<(int)((total + 255) / 256), 256, 0, stream>>>(
        SRC, DST, exb, H2, OUT, 5, total);
  }

  // ================= log_softmax =================
  log_softmax32<<<(N_NODES + 7) / 8, 256, 0, stream>>>(OUT, N_NODES);
}